// ImageDecoder_62156766708020
// MI455X (gfx1250) — compile-verified
//
#include <hip/hip_runtime.h>
#include <math.h>

#define HH   128
#define NPIX (HH * HH)
#define PI_F 3.14159265358979323846f

__global__ __launch_bounds__(256) void ImageDecoder_kernel(
    const float* __restrict__ W,
    const float* __restrict__ ngauss,
    const float* __restrict__ nunif,
    float* __restrict__ out)
{
    __shared__ float4 bufG[2][256];
    __shared__ float4 bufU[2][256];
    __shared__ float  smn[8], smx[8];

    const int b = blockIdx.x;
    const int t = threadIdx.x;

    // ---- per-image weights (uniform across block -> scalar loads) ----
    const float* Wb = W + (size_t)b * 16;
    const float w0  = Wb[0],  w1  = Wb[1],  w2  = Wb[2],  w3  = Wb[3];
    const float w4  = Wb[4],  w5  = Wb[5],  w6  = Wb[6],  w7  = Wb[7];
    const float w8  = Wb[8],  w9  = Wb[9],  w10 = Wb[10], w11 = Wb[11];
    const float w12 = Wb[12], w13 = Wb[13], w14 = Wb[14], w15 = Wb[15];

    // ---- per-image scalars ----
    const float disk0 = w0 * 6.0f;
    const float half  = floorf(fminf(fmaxf(fabsf(w1) * 8.0f + 2.0f, 2.0f), 12.0f));
    const float sqlo  = 64.0f - half, sqhi = 64.0f + half;
    const float a2    = w2 * (2.0f * PI_F / 128.0f);
    const float a4    = w4 * (PI_F / 256.0f);
    const float off5  = w5 * 10.0f;
    const float off7  = truncf(w7 * 5.0f);
    const float bxy   = 64.0f + off7;                 // bx == by
    const float sg    = 4.0f + fabsf(w7) * 5.0f;
    const float n2s2  = -1.0f / (2.0f * sg * sg);
    const float r8    = w8 * 10.0f;
    const float cbs   = floorf(fminf(fmaxf(fabsf(w9) * 8.0f + 2.0f, 2.0f), 16.0f));
    const float ang   = w10 * PI_F;
    const float ca    = __cosf(ang), sa = __sinf(ang);
    const float w114  = w11 * 4.0f;
    const float gsc   = 0.2f * w12;
    const float usc   = 0.2f * w13;
    const float con   = 1.0f + w14;
    const float ccon  = 0.5f - 0.5f * con;            // (v-0.5)*con+0.5 == v*con + ccon
    const bool  flip  = (w15 > 0.0f);

    // ---- j-dependent hoists: thread t owns columns j0..j0+3 forever ----
    const int j0 = (t & 31) * 4;
    float dy[4], ey[4], roty[4], base[4], a4fj[4], sq05[4], pj3[4];
#pragma unroll
    for (int c = 0; c < 4; ++c) {
        const float fj = (float)(j0 + c);
        dy[c]   = fj - 64.0f;
        const float s3 = 0.5f * __sinf(w3 * (2.0f * PI_F / 128.0f) * fj);
        const float s5 = 0.5f * __sinf((fj + off5) * (PI_F / 128.0f));
        base[c] = s3 + s5 + w6 - 0.5f * usc;          // fold brightness + uniform-noise bias
        const float dby = fj - bxy;
        ey[c]   = __expf(dby * dby * n2s2);
        roty[c] = sa * dy[c];
        a4fj[c] = a4 * fj;
        sq05[c] = ((fj >= sqlo) & (fj < sqhi)) ? 0.5f : 0.0f;
        const float cbj  = floorf(fj / cbs);
        const float parj = cbj - floorf(cbj * 0.5f) * 2.0f;   // cbj mod 2
        pj3[c]  = 0.3f * parj;
    }

    const int      wid   = t >> 5;  // wave id; also sub-row index
    const uint64_t gbase = (uint64_t)(uintptr_t)(ngauss + (size_t)b * NPIX);
    const uint64_t ubase = (uint64_t)(uintptr_t)(nunif  + (size_t)b * NPIX);

    // async double-buffered noise staging: 2 x b128 per thread per chunk
    auto issue = [&](int k) {
        const uint64_t ga = gbase + (uint64_t)((k * 1024 + t * 4) * 4);
        const uint64_t ua = ubase + (uint64_t)((k * 1024 + t * 4) * 4);
        const uint32_t lg = (uint32_t)(uintptr_t)&bufG[k & 1][t];
        const uint32_t lu = (uint32_t)(uintptr_t)&bufU[k & 1][t];
        asm volatile("global_load_async_to_lds_b128 %0, %1, off"
                     :: "v"(lg), "v"(ga) : "memory");
        asm volatile("global_load_async_to_lds_b128 %0, %1, off"
                     :: "v"(lu), "v"(ua) : "memory");
    };

    issue(0);

    float img[64];
    float mn = 1e30f, mx = -1e30f;

    // atan minimax (deg-11, |err| ~ 2e-6 rad on [0,1])
    const float C1 = 0.99997726f, C3 = -0.33262347f, C5 = 0.19354346f;
    const float C7 = -0.11643287f, C9 = 0.05265332f, C11 = -0.01172120f;

#pragma unroll
    for (int k = 0; k < 16; ++k) {
        issue(k < 15 ? k + 1 : 15);   // k==15 re-issues same data: benign, keeps wait count uniform
        asm volatile("s_wait_asynccnt 0x2" ::: "memory"); // chunk k resident (in-order retire)

        const float4 gg = bufG[k & 1][t];
        const float4 uu = bufU[k & 1][t];
        const float gA[4] = {gg.x, gg.y, gg.z, gg.w};
        const float uA[4] = {uu.x, uu.y, uu.z, uu.w};

        // i-dependent hoists
        const float fi    = (float)(k * 8 + wid);
        const float dx    = fi - 64.0f;
        const float dx2   = dx * dx;
        const float s2    = 0.5f * __sinf(a2 * fi);
        const float dbx   = fi - bxy;
        const float exi   = __expf(dbx * dbx * n2s2);
        const float rotx  = ca * dx;
        const float a4fi  = a4 * fi;
        const float sqi_f = ((fi >= sqlo) & (fi < sqhi)) ? 1.0f : 0.0f;
        const float cbi   = floorf(fi / cbs);
        const float pari  = cbi - floorf(cbi * 0.5f) * 2.0f;  // cbi mod 2
        const float pa3   = 0.3f * pari;
        const float om2A  = 1.0f - 2.0f * pari;               // checker: 0.3*(A^B) = pa3 + pj3*(1-2A)

#pragma unroll
        for (int c = 0; c < 4; ++c) {
            const float r = __builtin_amdgcn_sqrtf(fmaf(dy[c], dy[c], dx2));
            float v = fminf(fmaxf(disk0 - r, 0.0f), 1.0f);    // disk (med3)
            v = fmaf(sqi_f, sq05[c], v);                      // square
            v += base[c] + s2;                                // axis sinusoids + consts
            v = fmaf(0.5f, __sinf(a4fi + a4fj[c]), v);        // diag sinusoid
            v = fmaf(exi, ey[c], v);                          // gaussian blob
            const float rr = r - r8;
            v += (rr * rr < 10.0f) ? 1.0f : 0.0f;             // ring
            v += fmaf(pj3[c], om2A, pa3);                     // checkerboard
            const float rl = rotx + roty[c];
            v += (fabsf(rl) < 3.0f) ? 0.6f : 0.0f;            // rotated line

            // fast atan2(dy, dx) via poly (rcp on idle trans pipe)
            const float ax = fabsf(dx), ay = fabsf(dy[c]);
            const float h2 = fmaxf(ax, ay), l2 = fminf(ax, ay);
            const float zq = (h2 == 0.0f) ? 0.0f : l2 * __builtin_amdgcn_rcpf(h2);
            const float z2 = zq * zq;
            float p = zq * fmaf(z2, fmaf(z2, fmaf(z2, fmaf(z2, fmaf(z2, C11, C9), C7), C5), C3), C1);
            p = (ay > ax)  ? 1.57079632679f - p : p;
            p = (dx < 0.0f) ? PI_F - p : p;
            const float th = copysignf(p, dy[c]);
            v = fmaf(0.5f, __sinf(th * w114), v);             // angular sinusoid

            v = fmaf(gA[c], gsc, v);                          // gaussian noise
            v = fmaf(uA[c], usc, v);                          // uniform noise (bias folded)
            v = fmaf(v, con, ccon);                           // contrast
            img[k * 4 + c] = v;
            mn = fminf(mn, v);
            mx = fmaxf(mx, v);
        }
    }

    // ---- block-wide min/max: wave32 shuffle tree, then cross-wave via LDS ----
#pragma unroll
    for (int o = 16; o > 0; o >>= 1) {
        mn = fminf(mn, __shfl_xor(mn, o, 32));
        mx = fmaxf(mx, __shfl_xor(mx, o, 32));
    }
    if ((t & 31) == 0) { smn[wid] = mn; smx[wid] = mx; }
    __syncthreads();
    mn = smn[0]; mx = smx[0];
#pragma unroll
    for (int wv = 1; wv < 8; ++wv) {
        mn = fminf(mn, smn[wv]);
        mx = fmaxf(mx, smx[wv]);
    }

    // inversion folded into normalization: (1-v - (1-mx))/(range) == (mx - v)/(range)
    const float inv = __builtin_amdgcn_rcpf(mx - mn + 1e-8f);
    const float sub = flip ? mx   : mn;
    const float scl = flip ? -inv : inv;

    float* ob = out + (size_t)b * NPIX;
#pragma unroll
    for (int k = 0; k < 16; ++k) {
        float4 o4;
        o4.x = (img[k * 4 + 0] - sub) * scl;
        o4.y = (img[k * 4 + 1] - sub) * scl;
        o4.z = (img[k * 4 + 2] - sub) * scl;
        o4.w = (img[k * 4 + 3] - sub) * scl;
        *(float4*)(ob + k * 1024 + t * 4) = o4;   // coalesced b128 store
    }
}

extern "C" void kernel_launch(void* const* d_in, const int* in_sizes, int n_in,
                              void* d_out, int out_size, void* d_ws, size_t ws_size,
                              hipStream_t stream) {
    const float* W  = (const float*)d_in[0];
    const float* ng = (const float*)d_in[1];
    const float* nu = (const float*)d_in[2];
    float*       o  = (float*)d_out;
    const int B = in_sizes[0] / 16;   // W is (B,16)
    ImageDecoder_kernel<<<B, 256, 0, stream>>>(W, ng, nu, o);
    (void)n_in; (void)out_size; (void)d_ws; (void)ws_size;
}